// PoincareHead_54485955117770
// MI455X (gfx1250) — compile-verified
//
#include <hip/hip_runtime.h>
#include <hip/hip_bf16.h>
#include <math.h>

typedef float v2f __attribute__((ext_vector_type(2)));
typedef float v8f __attribute__((ext_vector_type(8)));

#define B_ROWS 8192
#define C_ROWS 4096
#define DIM    512
#define EPSF   1e-6f
#define MAXN   0.999f          // 1.0 - 1e-3

// ---------------------------------------------------------------------------
// Kernel 1: row-wise expmap0 (optional) + clip-to-ball + squared-norm output.
// One wave32 per row; each lane handles 4 float4 = 16 floats of the 512-dim row.
// ---------------------------------------------------------------------------
__global__ __launch_bounds__(256) void poincare_rowprep(
    const float* __restrict__ src, const float* __restrict__ log_c,
    float* __restrict__ dst, float* __restrict__ sq_out,
    int nrows, int apply_expmap)
{
    const int gtid = blockIdx.x * blockDim.x + threadIdx.x;
    const int wave = gtid >> 5;
    const int lane = threadIdx.x & 31;
    if (wave >= nrows) return;

    const float4* row = (const float4*)(src + (size_t)wave * DIM);
    float4 v[4];
    float ss = 0.f;
#pragma unroll
    for (int i = 0; i < 4; ++i) {
        v[i] = row[lane + 32 * i];
        ss += v[i].x * v[i].x + v[i].y * v[i].y + v[i].z * v[i].z + v[i].w * v[i].w;
    }
    // wave32 butterfly reduction
#pragma unroll
    for (int off = 16; off > 0; off >>= 1)
        ss += __shfl_xor(ss, off, 32);

    const float c  = expf(log_c[0]);
    const float sc = sqrtf(c);

    const float vn = fmaxf(sqrtf(ss), EPSF);
    float coeff = 1.0f;
    if (apply_expmap)
        coeff = tanhf(sc * vn) / (sc * vn);

    const float tn     = fmaxf(coeff * vn, EPSF);
    const float factor = fminf(MAXN / tn, 1.0f);
    const float s      = coeff * factor;          // total per-row scale

    float4* out = (float4*)(dst + (size_t)wave * DIM);
#pragma unroll
    for (int i = 0; i < 4; ++i) {
        float4 o;
        o.x = v[i].x * s; o.y = v[i].y * s; o.z = v[i].z * s; o.w = v[i].w * s;
        out[lane + 32 * i] = o;
    }
    if (lane == 0)
        sq_out[wave] = s * s * ss;               // raw ||row||^2 after map
}

// ---------------------------------------------------------------------------
// Kernel 2: FP32 WMMA GEMM (xy = x_h @ cen_h^T) fused with the Poincare
// distance epilogue.  Block = 256 threads = 8 waves (2 x 4), block tile
// 64(M) x 128(N), K-tile 32.  Double-buffered LDS filled with
// GLOBAL_LOAD_ASYNC_TO_LDS_B128 (ASYNCcnt) so tile t+1 loads overlap the
// WMMA work on tile t.  Each wave owns a 32x32 output tile = 2x2 chains of
// V_WMMA_F32_16X16X4_F32.
// ---------------------------------------------------------------------------
#define M_BLK 64
#define N_BLK 128
#define K_TILE 32
#define LDA (K_TILE + 2)                 // +2 floats keeps v2f LDS reads 8B aligned
#define BUF_FLTS (M_BLK * LDA + N_BLK * LDA)   // 6528 floats per buffer
#define NITER (DIM / K_TILE)             // 16

__global__ __launch_bounds__(256) void poincare_gemm_dist(
    const float* __restrict__ x_h, const float* __restrict__ cen_h,
    const float* __restrict__ log_c,
    const float* __restrict__ x2_raw, const float* __restrict__ y2_raw,
    float* __restrict__ dists)
{
    __shared__ __attribute__((aligned(16))) float smem[2 * BUF_FLTS]; // 52224 B

    const int tid  = threadIdx.x;
    const int wid  = tid >> 5;
    const int lane = tid & 31;
    const int wm   = wid >> 2;          // 0..1  (M direction)
    const int wn   = wid & 3;           // 0..3  (N direction)
    const int half = lane >> 4;         // 0: K0/K1 lanes, 1: K2/K3 lanes
    const int i16  = lane & 15;

    const int m0 = blockIdx.y * M_BLK;
    const int n0 = blockIdx.x * N_BLK;

    // Issue one K-tile of async global->LDS loads (6 x b128 per thread).
    auto issueTile = [&](int kt, int buf) {
        const unsigned abase = (unsigned)(size_t)&smem[buf * BUF_FLTS];
        const unsigned bbase = (unsigned)(size_t)&smem[buf * BUF_FLTS + M_BLK * LDA];
        // A tile: 64 rows x 32 floats = 512 float4 -> 2 per thread
#pragma unroll
        for (int j = 0; j < 2; ++j) {
            const int i  = tid + 256 * j;
            const int r  = i >> 3;
            const int c4 = i & 7;
            const unsigned l = abase + (unsigned)((r * LDA + c4 * 4) * 4);
            const unsigned g = (unsigned)((((m0 + r) * DIM) + kt + c4 * 4) * 4);
            asm volatile("global_load_async_to_lds_b128 %0, %1, %2"
                         :
                         : "v"(l), "v"(g), "s"(x_h)
                         : "memory");
        }
        // B tile: 128 rows x 32 floats = 1024 float4 -> 4 per thread
#pragma unroll
        for (int j = 0; j < 4; ++j) {
            const int i  = tid + 256 * j;
            const int r  = i >> 3;
            const int c4 = i & 7;
            const unsigned l = bbase + (unsigned)((r * LDA + c4 * 4) * 4);
            const unsigned g = (unsigned)((((n0 + r) * DIM) + kt + c4 * 4) * 4);
            asm volatile("global_load_async_to_lds_b128 %0, %1, %2"
                         :
                         : "v"(l), "v"(g), "s"(cen_h)
                         : "memory");
        }
    };

    v8f acc[2][2];
#pragma unroll
    for (int ti = 0; ti < 2; ++ti)
#pragma unroll
        for (int tj = 0; tj < 2; ++tj)
            acc[ti][tj] = (v8f)0.0f;

    issueTile(0, 0);                       // prologue: fill buffer 0

    for (int it = 0; it < NITER; ++it) {
        // Our async writes for the current buffer have landed...
        asm volatile("s_wait_asynccnt 0x0" ::: "memory");
        // ...and after the barrier everyone's have (also orders the previous
        // iteration's LDS reads before the next buffer's async writes).
        __syncthreads();
        if (it + 1 < NITER)
            issueTile((it + 1) * K_TILE, (it + 1) & 1);

        const float* As = &smem[(it & 1) * BUF_FLTS];
        const float* Bs = As + M_BLK * LDA;

#pragma unroll
        for (int kk = 0; kk < K_TILE; kk += 4) {
            // ISA 16x4 f32 A layout: lanes 0-15 hold K=kk,kk+1; lanes 16-31 K=kk+2,kk+3
            v2f a[2], b[2];
#pragma unroll
            for (int ti = 0; ti < 2; ++ti)
                a[ti] = *(const v2f*)&As[(wm * 32 + ti * 16 + i16) * LDA + kk + 2 * half];
#pragma unroll
            for (int tj = 0; tj < 2; ++tj)
                b[tj] = *(const v2f*)&Bs[(wn * 32 + tj * 16 + i16) * LDA + kk + 2 * half];
#pragma unroll
            for (int ti = 0; ti < 2; ++ti)
#pragma unroll
                for (int tj = 0; tj < 2; ++tj)
                    acc[ti][tj] = __builtin_amdgcn_wmma_f32_16x16x4_f32(
                        false, a[ti], false, b[tj], (short)0, acc[ti][tj],
                        false, false);
        }
    }

    // ---------------- epilogue: Poincare distance ----------------
    const float c   = expf(log_c[0]);
    const float rsc = 1.0f / sqrtf(c);
    const float M2  = MAXN * MAXN;

#pragma unroll
    for (int ti = 0; ti < 2; ++ti) {
#pragma unroll
        for (int tj = 0; tj < 2; ++tj) {
            const int mrow0 = m0 + wm * 32 + ti * 16;
            const int ncol  = n0 + wn * 32 + tj * 16 + i16;
            const float y2r = y2_raw[ncol];
            const float dy  = 1.0f - c * fminf(y2r, M2);
#pragma unroll
            for (int r = 0; r < 8; ++r) {
                const int m = mrow0 + r + half * 8;   // C/D layout: VGPR r -> row r / r+8
                const float x2r   = x2_raw[m];
                const float dx    = 1.0f - c * fminf(x2r, M2);
                const float diff2 = x2r + y2r - 2.0f * acc[ti][tj][r];
                const float denom = fmaxf(dx * dy, EPSF);
                const float arg   = fmaxf(1.0f + (2.0f * c) * diff2 / denom, 1.0f + EPSF);
                // arccosh(arg) = log(arg + sqrt(arg^2 - 1))
                const float dist  = logf(arg + sqrtf(arg * arg - 1.0f)) * rsc;
                dists[(size_t)m * C_ROWS + ncol] = dist;
            }
        }
    }
}

// ---------------------------------------------------------------------------
extern "C" void kernel_launch(void* const* d_in, const int* in_sizes, int n_in,
                              void* d_out, int out_size, void* d_ws, size_t ws_size,
                              hipStream_t stream)
{
    const float* emb       = (const float*)d_in[0];   // (8192, 512)
    const float* log_c     = (const float*)d_in[1];   // scalar
    const float* centroids = (const float*)d_in[2];   // (4096, 512)

    float* out   = (float*)d_out;
    float* dists = out;                                   // (B, C)
    float* x_h   = out + (size_t)B_ROWS * C_ROWS;         // (B, D)
    float* cen_h = x_h + (size_t)B_ROWS * DIM;            // (C, D)

    float* x2 = (float*)d_ws;          // B raw squared norms
    float* y2 = x2 + B_ROWS;           // C raw squared norms

    // 8 waves (rows) per 256-thread block
    poincare_rowprep<<<B_ROWS / 8, 256, 0, stream>>>(emb, log_c, x_h, x2,
                                                     B_ROWS, 1);
    poincare_rowprep<<<C_ROWS / 8, 256, 0, stream>>>(centroids, log_c, cen_h, y2,
                                                     C_ROWS, 0);

    dim3 grid(C_ROWS / N_BLK, B_ROWS / M_BLK);   // (32, 128)
    poincare_gemm_dist<<<grid, 256, 0, stream>>>(x_h, cen_h, log_c, x2, y2, dists);
}